// LayoutLMv3KVPModel_14199161881074
// MI455X (gfx1250) — compile-verified
//
#include <hip/hip_runtime.h>
#include <math.h>

typedef __attribute__((ext_vector_type(2))) float v2f;
typedef __attribute__((ext_vector_type(4))) float v4f;
typedef __attribute__((ext_vector_type(8))) float v8f;

#define BB   32
#define SS   512
#define HH   768
#define MAXC 128
#define FEPS 1e-8f
#define KC   32          // K-chunk staged in LDS per block
#define LPAD 34          // LDS row stride (even -> 8B-aligned ds_load_b64)

// ---------------------------------------------------------------------------
// Kernel 1: softmax + argmax -> masked key/val confidences
// ---------------------------------------------------------------------------
__global__ void conf_kernel(const float* __restrict__ logits,   // [B,S,3]
                            const int*   __restrict__ mask,     // [B,S]
                            float* __restrict__ kc,             // [B,S]
                            float* __restrict__ vc) {           // [B,S]
    int gid = blockIdx.x * blockDim.x + threadIdx.x;            // B*S
    float l0 = logits[gid * 3 + 0];
    float l1 = logits[gid * 3 + 1];
    float l2 = logits[gid * 3 + 2];
    int pred = 0; float mx = l0;
    if (l1 > mx) { mx = l1; pred = 1; }      // strict '>' = argmax first-occurrence
    if (l2 > mx) { mx = l2; pred = 2; }
    float e0 = __expf(l0 - mx), e1 = __expf(l1 - mx), e2 = __expf(l2 - mx);
    float inv = 1.0f / (e0 + e1 + e2);
    bool valid = (mask[gid] == 1);
    kc[gid] = (pred == 1 && valid) ? e1 * inv : -INFINITY;
    vc[gid] = (pred == 2 && valid) ? e2 * inv : -INFINITY;
}

// ---------------------------------------------------------------------------
// Kernel 2: stable top-128 of 512 via bitonic sort (desc value, asc index)
// ---------------------------------------------------------------------------
__global__ void topk_kernel(const float* __restrict__ conf,     // [B,S]
                            int*   __restrict__ idx_out,        // [B,MAXC] ws
                            float* __restrict__ valid_out,      // [B,MAXC] d_out
                            float* __restrict__ idxf_out) {     // [B,MAXC] d_out
    __shared__ float sv[SS];
    __shared__ int   si[SS];
    int b = blockIdx.x, t = threadIdx.x;
    sv[t] = conf[b * SS + t];
    si[t] = t;
    __syncthreads();
    for (int k = 2; k <= SS; k <<= 1) {
        for (int j = k >> 1; j > 0; j >>= 1) {
            int ixj = t ^ j;
            if (ixj > t) {
                float v0 = sv[t], v1 = sv[ixj];
                int   i0 = si[t], i1 = si[ixj];
                bool gt = (v0 > v1) || (v0 == v1 && i0 < i1);   // stable desc order
                bool desc = ((t & k) == 0);
                if (desc ? !gt : gt) {
                    sv[t] = v1; si[t] = i1; sv[ixj] = v0; si[ixj] = i0;
                }
            }
            __syncthreads();
        }
    }
    if (t < MAXC) {
        idx_out[b * MAXC + t]   = si[t];
        valid_out[b * MAXC + t] = (sv[t] > -INFINITY) ? 1.0f : 0.0f;  // isfinite
        idxf_out[b * MAXC + t]  = (float)si[t];
    }
}

// ---------------------------------------------------------------------------
// Kernel 3: (optionally gathered) GEMM  out[b] = A[b] @ W + bias
// One block (8 waves, 256 threads) owns one (batch, 64-col strip); the 8
// waves cover the 8 M-tiles and share the W strip staged through LDS in
// transposed layout, so each B fragment is one contiguous ds_load_b64.
// ---------------------------------------------------------------------------
__global__ void __launch_bounds__(256)
wmma_gemm_f32(const float* __restrict__ A,
              const int*   __restrict__ gidx,   // [B,M] or null
              long a_bstride, int lda,
              const float* __restrict__ W,      // [K,N] row-major
              const float* __restrict__ bias,   // [N] or null
              float* __restrict__ out,          // [B,M,N]
              int M, int N, int K) {
    __shared__ float ldsWT[64 * LPAD];          // [col 0..63][k 0..KC-1], stride LPAD

    int nstrips = N >> 6;
    int b  = blockIdx.x / nstrips;
    int ns = blockIdx.x % nstrips;
    int mt = threadIdx.x >> 5;                  // wave id = M-tile (8 waves, M=128)
    int lane = threadIdx.x & 31;
    int half = lane >> 4;                       // 0: K+0/1, 1: K+2/3
    int l16  = lane & 15;

    int mrow = mt * 16 + l16;
    const float* arow;
    if (gidx) arow = A + b * a_bstride + (long)gidx[b * M + mrow] * lda;
    else      arow = A + b * a_bstride + (long)mrow * lda;

    int ncol0 = ns * 64;                        // strip's first global column
    // staging coords: thread covers 2 float4s of the 32x64 chunk
    int sr0 = threadIdx.x >> 4;                 // rows 0..15  (idx 0..255)
    int sr1 = (threadIdx.x + 256) >> 4;         // rows 16..31
    int scq = threadIdx.x & 15;                 // float4 column group 0..15

    v8f c0 = {}, c1 = {}, c2 = {}, c3 = {};
    for (int kb = 0; kb < K; kb += KC) {
        // ---- cooperative stage: W[kb..kb+31][ncol0..ncol0+63] -> ldsWT ----
        {
            v4f w0 = *(const v4f*)(W + (long)(kb + sr0) * N + ncol0 + scq * 4);
            v4f w1 = *(const v4f*)(W + (long)(kb + sr1) * N + ncol0 + scq * 4);
            int cbase = scq * 4;
            ldsWT[(cbase + 0) * LPAD + sr0] = w0.x;
            ldsWT[(cbase + 1) * LPAD + sr0] = w0.y;
            ldsWT[(cbase + 2) * LPAD + sr0] = w0.z;
            ldsWT[(cbase + 3) * LPAD + sr0] = w0.w;
            ldsWT[(cbase + 0) * LPAD + sr1] = w1.x;
            ldsWT[(cbase + 1) * LPAD + sr1] = w1.y;
            ldsWT[(cbase + 2) * LPAD + sr1] = w1.z;
            ldsWT[(cbase + 3) * LPAD + sr1] = w1.w;
        }
        __syncthreads();
        // ---- consume: 8 K-steps of 4, A fragment reused over 4 N-tiles ----
        #pragma unroll
        for (int kk = 0; kk < KC; kk += 4) {
            int ka = kk + half * 2;             // lane-local K offset (even)
            v2f a = *(const v2f*)(arow + kb + ka);
            v2f b0 = *(const v2f*)(&ldsWT[(l16)      * LPAD + ka]);
            v2f b1 = *(const v2f*)(&ldsWT[(l16 + 16) * LPAD + ka]);
            v2f b2 = *(const v2f*)(&ldsWT[(l16 + 32) * LPAD + ka]);
            v2f b3 = *(const v2f*)(&ldsWT[(l16 + 48) * LPAD + ka]);
            c0 = __builtin_amdgcn_wmma_f32_16x16x4_f32(false, a, false, b0, (short)0, c0, false, false);
            c1 = __builtin_amdgcn_wmma_f32_16x16x4_f32(false, a, false, b1, (short)0, c1, false, false);
            c2 = __builtin_amdgcn_wmma_f32_16x16x4_f32(false, a, false, b2, (short)0, c2, false, false);
            c3 = __builtin_amdgcn_wmma_f32_16x16x4_f32(false, a, false, b3, (short)0, c3, false, false);
        }
        __syncthreads();
    }
    int cA = ncol0 + l16;
    float bn0 = bias ? bias[cA]      : 0.0f;
    float bn1 = bias ? bias[cA + 16] : 0.0f;
    float bn2 = bias ? bias[cA + 32] : 0.0f;
    float bn3 = bias ? bias[cA + 48] : 0.0f;
    long obase = (long)b * M * N;
    #pragma unroll
    for (int r = 0; r < 8; ++r) {
        int row = mt * 16 + r + half * 8;
        long o = obase + (long)row * N + cA;
        out[o]      = c0[r] + bn0;
        out[o + 16] = c1[r] + bn1;
        out[o + 32] = c2[r] + bn2;
        out[o + 48] = c3[r] + bn3;
    }
}

// ---------------------------------------------------------------------------
// Kernel 4: biaffine  out[b] = A[b] @ Bm[b]^T + bbil   (A:[M,K], Bm:[N,K])
// Both operands are K-contiguous -> all fragments are global_load_b64.
// ---------------------------------------------------------------------------
__global__ void wmma_abt_f32(const float* __restrict__ A,       // [B,M,K]
                             const float* __restrict__ Bm,      // [B,N,K]
                             const float* __restrict__ bbil,    // scalar
                             float* __restrict__ out,           // [B,M,N]
                             int M, int N, int K) {
    int wave = (blockIdx.x * blockDim.x + threadIdx.x) >> 5;
    int lane = threadIdx.x & 31;
    int mtiles  = M >> 4;
    int nstrips = N >> 6;
    int perB = mtiles * nstrips;
    int b  = wave / perB;
    int t  = wave % perB;
    int mt = t / nstrips;
    int ns = t % nstrips;

    int half = lane >> 4;
    int l16  = lane & 15;

    const float* arow = A  + (long)b * M * K + (long)(mt * 16 + l16) * K;
    int n0 = ns * 64 + l16;
    const float* v0 = Bm + (long)b * N * K + (long)(n0)      * K;
    const float* v1 = Bm + (long)b * N * K + (long)(n0 + 16) * K;
    const float* v2 = Bm + (long)b * N * K + (long)(n0 + 32) * K;
    const float* v3 = Bm + (long)b * N * K + (long)(n0 + 48) * K;

    v8f c0 = {}, c1 = {}, c2 = {}, c3 = {};
    for (int k = 0; k < K; k += 4) {
        int ka = k + half * 2;                  // even -> 8B-aligned b64 loads
        v2f a  = *(const v2f*)(arow + ka);
        v2f b0 = *(const v2f*)(v0 + ka);
        v2f b1 = *(const v2f*)(v1 + ka);
        v2f b2 = *(const v2f*)(v2 + ka);
        v2f b3 = *(const v2f*)(v3 + ka);
        c0 = __builtin_amdgcn_wmma_f32_16x16x4_f32(false, a, false, b0, (short)0, c0, false, false);
        c1 = __builtin_amdgcn_wmma_f32_16x16x4_f32(false, a, false, b1, (short)0, c1, false, false);
        c2 = __builtin_amdgcn_wmma_f32_16x16x4_f32(false, a, false, b2, (short)0, c2, false, false);
        c3 = __builtin_amdgcn_wmma_f32_16x16x4_f32(false, a, false, b3, (short)0, c3, false, false);
    }
    float bb = bbil[0];
    long obase = (long)b * M * N;
    #pragma unroll
    for (int r = 0; r < 8; ++r) {
        int row = mt * 16 + r + half * 8;
        long o = obase + (long)row * N + ns * 64 + l16;
        out[o]      = c0[r] + bb;
        out[o + 16] = c1[r] + bb;
        out[o + 32] = c2[r] + bb;
        out[o + 48] = c3[r] + bb;
    }
}

// ---------------------------------------------------------------------------
// Kernel 5: spatial features + pair MLPs (weights staged in LDS)
// ---------------------------------------------------------------------------
__global__ void score_kernel(const float* __restrict__ bboxes,  // [B,S,4]
                             const int*   __restrict__ kidx,    // [B,MAXC]
                             const int*   __restrict__ vidx,    // [B,MAXC]
                             const float* __restrict__ biaf,    // [B,MAXC,MAXC]
                             const float* __restrict__ Ws1, const float* __restrict__ bs1,
                             const float* __restrict__ Ws2, const float* __restrict__ bs2,
                             const float* __restrict__ Wf1, const float* __restrict__ bf1,
                             const float* __restrict__ Wf2, const float* __restrict__ bf2,
                             float* __restrict__ scores) {      // [B,MAXC,MAXC]
    __shared__ float sWs1[8 * 64], sbs1[64];
    __shared__ float sWs2[64 * 32], sbs2[32];
    __shared__ float sWf1[33 * 16], sbf1[16];
    __shared__ float sWf2[16], sbf2[1];
    for (int i = threadIdx.x; i < 8 * 64;  i += blockDim.x) sWs1[i] = Ws1[i];
    for (int i = threadIdx.x; i < 64;      i += blockDim.x) sbs1[i] = bs1[i];
    for (int i = threadIdx.x; i < 64 * 32; i += blockDim.x) sWs2[i] = Ws2[i];
    for (int i = threadIdx.x; i < 32;      i += blockDim.x) sbs2[i] = bs2[i];
    for (int i = threadIdx.x; i < 33 * 16; i += blockDim.x) sWf1[i] = Wf1[i];
    for (int i = threadIdx.x; i < 16;      i += blockDim.x) sbf1[i] = bf1[i];
    for (int i = threadIdx.x; i < 16;      i += blockDim.x) sWf2[i] = Wf2[i];
    if (threadIdx.x == 0) sbf2[0] = bf2[0];
    __syncthreads();

    int gid = blockIdx.x * blockDim.x + threadIdx.x;   // b*MAXC*MAXC + i*MAXC + j
    int b  = gid >> 14;
    int ij = gid & 16383;
    int i  = ij >> 7;
    int j  = ij & 127;

    const float* kb = bboxes + ((long)b * SS + kidx[b * MAXC + i]) * 4;
    const float* vb = bboxes + ((long)b * SS + vidx[b * MAXC + j]) * 4;
    float kx1 = kb[0], ky1 = kb[1], kx2 = kb[2], ky2 = kb[3];
    float vx1 = vb[0], vy1 = vb[1], vx2 = vb[2], vy2 = vb[3];
    float dx = (vx1 + vx2) * 0.5f - (kx1 + kx2) * 0.5f;
    float dy = (vy1 + vy2) * 0.5f - (ky1 + ky2) * 0.5f;
    float dist  = sqrtf(dx * dx + dy * dy + FEPS);
    float angle = atan2f(dy, dx);
    float kh = ky2 - ky1, kw = kx2 - kx1, vh = vy2 - vy1, vw = vx2 - vx1;
    float h_ov = fmaxf(fminf(ky2, vy2) - fmaxf(ky1, vy1), 0.0f);
    float h_al = h_ov / (fminf(kh, vh) + FEPS);
    float v_ov = fmaxf(fminf(kx2, vx2) - fmaxf(kx1, vx1), 0.0f);
    float v_al = v_ov / (fminf(kw, vw) + FEPS);
    float area = (vh * vw) / (kh * kw + FEPS);
    float aspc = (vw / (vh + FEPS)) / (kw / (kh + FEPS));
    float sf[8] = {dx, dy, dist, angle, h_al, v_al, area, aspc};

    float s1[64];
    #pragma unroll
    for (int n = 0; n < 64; ++n) {
        float acc = sbs1[n];
        #pragma unroll
        for (int p = 0; p < 8; ++p) acc += sf[p] * sWs1[p * 64 + n];
        s1[n] = fmaxf(acc, 0.0f);
    }
    float s2[32];
    #pragma unroll
    for (int n = 0; n < 32; ++n) {
        float acc = sbs2[n];
        for (int p = 0; p < 64; ++p) acc += s1[p] * sWs2[p * 32 + n];
        s2[n] = acc;
    }
    float c0 = biaf[gid];
    float sc = sbf2[0];
    #pragma unroll
    for (int n = 0; n < 16; ++n) {
        float acc = sbf1[n] + c0 * sWf1[n];
        #pragma unroll
        for (int p = 0; p < 32; ++p) acc += s2[p] * sWf1[(p + 1) * 16 + n];
        sc += fmaxf(acc, 0.0f) * sWf2[n];
    }
    scores[gid] = sc;
}

// ---------------------------------------------------------------------------
extern "C" void kernel_launch(void* const* d_in, const int* in_sizes, int n_in,
                              void* d_out, int out_size, void* d_ws, size_t ws_size,
                              hipStream_t stream) {
    const float* seq    = (const float*)d_in[0];
    const float* logits = (const float*)d_in[1];
    const float* bboxes = (const float*)d_in[2];
    const int*   amask  = (const int*)  d_in[3];
    const float* Wk  = (const float*)d_in[4];  const float* bk   = (const float*)d_in[5];
    const float* Wv  = (const float*)d_in[6];  const float* bv   = (const float*)d_in[7];
    const float* Wbil= (const float*)d_in[8];  const float* bbil = (const float*)d_in[9];
    const float* Ws1 = (const float*)d_in[10]; const float* bs1  = (const float*)d_in[11];
    const float* Ws2 = (const float*)d_in[12]; const float* bs2  = (const float*)d_in[13];
    const float* Wf1 = (const float*)d_in[14]; const float* bf1  = (const float*)d_in[15];
    const float* Wf2 = (const float*)d_in[16]; const float* bf2  = (const float*)d_in[17];

    // workspace carve-up (all 4-byte aligned; ~40 MB total)
    float* kc       = (float*)d_ws;                 // B*S
    float* vc       = kc + BB * SS;                 // B*S
    int*   kidx     = (int*)(vc + BB * SS);         // B*MAXC
    int*   vidx     = kidx + BB * MAXC;             // B*MAXC
    float* key_reps = (float*)(vidx + BB * MAXC);   // B*MAXC*H
    float* val_reps = key_reps + (long)BB * MAXC * HH;
    float* kW       = val_reps + (long)BB * MAXC * HH;
    float* biaf     = kW       + (long)BB * MAXC * HH;  // B*MAXC*MAXC

    // d_out layout: scores | key_valid | val_valid | kidx | vidx
    float* out_scores = (float*)d_out;
    float* out_kvalid = out_scores + BB * MAXC * MAXC;
    float* out_vvalid = out_kvalid + BB * MAXC;
    float* out_kidx   = out_vvalid + BB * MAXC;
    float* out_vidx   = out_kidx   + BB * MAXC;

    // 1) confidences
    conf_kernel<<<(BB * SS) / 256, 256, 0, stream>>>(logits, amask, kc, vc);

    // 2) top-128 per batch (keys, vals)
    topk_kernel<<<BB, SS, 0, stream>>>(kc, kidx, out_kvalid, out_kidx);
    topk_kernel<<<BB, SS, 0, stream>>>(vc, vidx, out_vvalid, out_vidx);

    // 3) gathered projections + 4) kW = key_reps @ Wbil
    {
        int blocks = BB * (HH / 64);                // 384: one block per (b, strip)
        wmma_gemm_f32<<<blocks, 256, 0, stream>>>(seq, kidx, (long)SS * HH, HH,
                                                  Wk, bk, key_reps, MAXC, HH, HH);
        wmma_gemm_f32<<<blocks, 256, 0, stream>>>(seq, vidx, (long)SS * HH, HH,
                                                  Wv, bv, val_reps, MAXC, HH, HH);
        wmma_gemm_f32<<<blocks, 256, 0, stream>>>(key_reps, nullptr, (long)MAXC * HH, HH,
                                                  Wbil, nullptr, kW, MAXC, HH, HH);
    }

    // 5) biaffine = kW @ val_reps^T + bbil
    {
        int waves = BB * (MAXC / 16) * (MAXC / 64); // 512
        int blocks = (waves * 32) / 256;            // 64
        wmma_abt_f32<<<blocks, 256, 0, stream>>>(kW, val_reps, bbil, biaf,
                                                 MAXC, MAXC, HH);
    }

    // 6) spatial MLPs + final scorer
    score_kernel<<<(BB * MAXC * MAXC) / 256, 256, 0, stream>>>(
        bboxes, kidx, vidx, biaf,
        Ws1, bs1, Ws2, bs2, Wf1, bf1, Wf2, bf2, out_scores);
}